// IF_43808666419805
// MI455X (gfx1250) — compile-verified
//
#include <hip/hip_runtime.h>

// Self-attention over a stack of L feature maps, fused flash-attention style.
// - Both GEMMs use V_WMMA_F32_16X16X4_F32 (fp32 precision, matrix pipe);
//   the [B,N,N] energy (4x42.5 MB) is never materialized.
// - One block (4 wave32) owns one (l, b, 16-query tile); waves split the
//   m-range 4 ways and merge online-softmax partials through LDS.
//   grid = L*B*144 = 1152 blocks = 4608 waves in ONE launch.
// - Two 16-column S tiles per online-softmax update: row reductions, alpha
//   exp and the O-rescale run once per 32 columns (halves softmax VALU).
// - Row reductions: DPP16 row_ror butterflies; raw v_max_num_f32 inline asm
//   avoids fmaxf's canonicalize self-max spam.
// - Softmax computed in base-2: log2(e) folded into the Q operands once,
//   exponentials are single v_exp_f32 (__builtin_amdgcn_exp2f).
// - Per-l partial outputs go to workspace; epilogue does
//   out = y + sum_l gamma_l * partial_l  (deterministic, no float atomics).
//
// Shapes: L=4 B=2 C=128 H=W=48 -> N=2304 (144 tiles of 16), CQ=16.

#define L_   4
#define B_   2
#define C_   128
#define HW_  48
#define N_   (HW_ * HW_)   // 2304
#define CQ_  16
#define NBLK (N_ / 16)     // 144
#define MPW  (NBLK / 4)    // 36 m-tiles per wave
#define LOG2E 1.4426950408889634f

typedef float v2f __attribute__((ext_vector_type(2)));
typedef float v8f __attribute__((ext_vector_type(8)));

// raw max (no NaN-canonicalize self-max emitted around it)
__device__ __forceinline__ float vmaxr(float a, float b) {
    float d;
    asm("v_max_num_f32 %0, %1, %2" : "=v"(d) : "v"(a), "v"(b));
    return d;
}

// ---- DPP16 row_ror butterfly reductions over a 16-lane group --------------
template <int CTRL>
__device__ __forceinline__ float row_rot(float x) {
    return __int_as_float(__builtin_amdgcn_update_dpp(
        0, __float_as_int(x), CTRL, 0xf, 0xf, true));
}
__device__ __forceinline__ float red_max16(float x) {
    x = vmaxr(x, row_rot<0x121>(x));   // row_ror:1
    x = vmaxr(x, row_rot<0x122>(x));   // row_ror:2
    x = vmaxr(x, row_rot<0x124>(x));   // row_ror:4
    x = vmaxr(x, row_rot<0x128>(x));   // row_ror:8
    return x;
}
__device__ __forceinline__ float red_sum16(float x) {
    x += row_rot<0x121>(x);
    x += row_rot<0x122>(x);
    x += row_rot<0x124>(x);
    x += row_rot<0x128>(x);
    return x;
}

// ---------------------------------------------------------------------------
// Projection q,k for all L elements.  qT layout: [L*B][n][cq] (A-operand
// friendly: contiguous cq).  k layout: [L*B][cq][m] (B-operand friendly).
// ---------------------------------------------------------------------------
__global__ __launch_bounds__(256) void proj_qk_kernel(
    const float* __restrict__ stack,
    const float* __restrict__ wq, const float* __restrict__ bq,
    const float* __restrict__ wk, const float* __restrict__ bk,
    float* __restrict__ qT, float* __restrict__ kK)
{
    __shared__ float swq[CQ_ * C_];
    __shared__ float swk[CQ_ * C_];
    for (int i = threadIdx.x; i < CQ_ * C_; i += blockDim.x) {
        swq[i] = wq[i];
        swk[i] = wk[i];
    }
    __syncthreads();

    const int g  = blockIdx.x * blockDim.x + threadIdx.x;   // 0 .. L*B*N-1
    const int n  = g % N_;
    const int lb = g / N_;                                  // l*B + b
    const float* xb = stack + (size_t)lb * C_ * N_;

    float aq[CQ_], ak[CQ_];
#pragma unroll
    for (int j = 0; j < CQ_; ++j) { aq[j] = 0.f; ak[j] = 0.f; }

    for (int c = 0; c < C_; ++c) {
        const float xv = xb[(size_t)c * N_ + n];
#pragma unroll
        for (int j = 0; j < CQ_; ++j) {
            aq[j] = fmaf(swq[j * C_ + c], xv, aq[j]);
            ak[j] = fmaf(swk[j * C_ + c], xv, ak[j]);
        }
    }

    float* qp = qT + ((size_t)lb * N_ + n) * CQ_;
#pragma unroll
    for (int j = 0; j < CQ_; ++j) qp[j] = aq[j] + bq[j];

    float* kp = kK + (size_t)lb * CQ_ * N_;
#pragma unroll
    for (int j = 0; j < CQ_; ++j) kp[(size_t)j * N_ + n] = ak[j] + bk[j];
}

// ---------------------------------------------------------------------------
// Value projection from y (shared across all l).  v layout: [b][c][m].
// ---------------------------------------------------------------------------
__global__ __launch_bounds__(256) void proj_v_kernel(
    const float* __restrict__ y,
    const float* __restrict__ wv, const float* __restrict__ bv,
    float* __restrict__ vV)
{
    const int g = blockIdx.x * blockDim.x + threadIdx.x;    // b*C*N elems
    const int n = g % N_;
    const int c = (g / N_) % C_;
    const int b = g / (C_ * N_);
    const float* yb = y + (size_t)b * C_ * N_;
    const float* wr = wv + (size_t)c * C_;
    float acc = bv[c];
    for (int cc = 0; cc < C_; ++cc)
        acc = fmaf(wr[cc], yb[(size_t)cc * N_ + n], acc);
    vV[g] = acc;   // (b*C + c)*N + n
}

// ---------------------------------------------------------------------------
// Fused flash attention.  Block = 4 wave32 = one (l, b, 16-query tile);
// wave w handles m-tiles [w*36, (w+1)*36), two tiles per softmax step.
// Writes softmax(Q^T K) V^T (per l, un-gamma'd) into pOut.
// ---------------------------------------------------------------------------
__global__ __launch_bounds__(128) void flash_attn_kernel(
    const float* __restrict__ qT, const float* __restrict__ kK,
    const float* __restrict__ vV, float* __restrict__ pOut)
{
    __shared__ float sO[4][16][132];        // partial O tiles (padded rows)
    __shared__ float sM[4][16];             // partial row max (log2 domain)
    __shared__ float sL[4][16];             // partial row sum
    __shared__ float sScale[4][16];         // merge rescale factors
    __shared__ float sLinv[16];             // merged 1/l*
    __shared__ float plds[4][2 * 16 * 18];  // per-wave P-tile bounce (2 tiles)

    const int lane = threadIdx.x & 31;
    const int wid  = threadIdx.x >> 5;
    const int l    = blockIdx.x / (B_ * NBLK);
    const int rem  = blockIdx.x % (B_ * NBLK);
    const int b    = rem / NBLK;
    const int n0   = (rem % NBLK) * 16;
    const int ci   = lane & 15;                 // column / row-M within tile
    const int half = lane >> 4;
    const int off  = half * 2;                  // K offset for A/B operands

    const float* qb = qT + (size_t)(l * B_ + b) * N_ * CQ_;
    const float* kb = kK + (size_t)(l * B_ + b) * CQ_ * N_;
    const float* vb = vV + (size_t)b * C_ * N_;
    float*       pb = pOut + (size_t)(l * B_ + b) * C_ * N_;
    float*       pl0 = plds[wid];
    float*       pl1 = plds[wid] + 16 * 18;

    // Q tile A-operands, pre-scaled by log2(e) so S is in base-2 domain.
    v2f aQ[4];
#pragma unroll
    for (int kk = 0; kk < 4; ++kk) {
        const float* p = qb + (size_t)(n0 + ci) * CQ_ + 4 * kk + off;
        aQ[kk].x = p[0] * LOG2E;
        aQ[kk].y = p[1] * LOG2E;
    }

    const v8f vzero = {};
    v8f O[8];
#pragma unroll
    for (int t = 0; t < 8; ++t) O[t] = vzero;
    float mrow[8], lrow[8];
#pragma unroll
    for (int r = 0; r < 8; ++r) { mrow[r] = -3.0e38f; lrow[r] = 0.f; }

    const int tile0 = wid * MPW;
    for (int it = 0; it < MPW / 2; ++it) {
        const int m0 = (tile0 + 2 * it) * 16;
        const int m1 = m0 + 16;

        // ---- S0,S1 = Q^T K for two adjacent 16-col tiles (8 wmma) ---------
        v8f S0 = vzero, S1 = vzero;
#pragma unroll
        for (int kk = 0; kk < 4; ++kk) {
            const float* kr0 = kb + (size_t)(4 * kk + off) * N_;
            const float* kr1 = kr0 + N_;
            v2f bK0, bK1;
            bK0.x = kr0[m0 + ci]; bK0.y = kr1[m0 + ci];
            bK1.x = kr0[m1 + ci]; bK1.y = kr1[m1 + ci];
            S0 = __builtin_amdgcn_wmma_f32_16x16x4_f32(
                     false, aQ[kk], false, bK0, (short)0, S0, false, false);
            S1 = __builtin_amdgcn_wmma_f32_16x16x4_f32(
                     false, aQ[kk], false, bK1, (short)0, S1, false, false);
        }

        // ---- one online-softmax update covering 32 columns ----------------
        v8f P0, P1;
#pragma unroll
        for (int r = 0; r < 8; ++r) {
            const float x0 = S0[r];
            const float x1 = S1[r];
            const float rm = red_max16(vmaxr(x0, x1));
            const float nm = vmaxr(mrow[r], rm);
            const float alpha = __builtin_amdgcn_exp2f(mrow[r] - nm);
            mrow[r] = nm;
            const float p0 = __builtin_amdgcn_exp2f(x0 - nm);
            const float p1 = __builtin_amdgcn_exp2f(x1 - nm);
            P0[r] = p0;
            P1[r] = p1;
            lrow[r] = lrow[r] * alpha + red_sum16(p0 + p1);
#pragma unroll
            for (int t = 0; t < 8; ++t) O[t][r] *= alpha;
        }

        // ---- re-shape P: D-layout -> LDS -> A-layout (in-wave DS order) ----
#pragma unroll
        for (int r = 0; r < 8; ++r) {
            pl0[(r + 8 * half) * 18 + ci] = P0[r];
            pl1[(r + 8 * half) * 18 + ci] = P1[r];
        }

        // ---- O += P * V^T : two tiles x 4 k-steps x 8 c-tiles (64 wmma) ---
#pragma unroll
        for (int kk = 0; kk < 4; ++kk) {
            v2f aP;
            aP.x = pl0[ci * 18 + 4 * kk + off];
            aP.y = pl0[ci * 18 + 4 * kk + off + 1];
#pragma unroll
            for (int t = 0; t < 8; ++t) {
                const float* vp = vb + (size_t)(16 * t + ci) * N_ + m0 + 4 * kk + off;
                v2f bV; bV.x = vp[0]; bV.y = vp[1];
                O[t] = __builtin_amdgcn_wmma_f32_16x16x4_f32(
                           false, aP, false, bV, (short)0, O[t], false, false);
            }
        }
#pragma unroll
        for (int kk = 0; kk < 4; ++kk) {
            v2f aP;
            aP.x = pl1[ci * 18 + 4 * kk + off];
            aP.y = pl1[ci * 18 + 4 * kk + off + 1];
#pragma unroll
            for (int t = 0; t < 8; ++t) {
                const float* vp = vb + (size_t)(16 * t + ci) * N_ + m1 + 4 * kk + off;
                v2f bV; bV.x = vp[0]; bV.y = vp[1];
                O[t] = __builtin_amdgcn_wmma_f32_16x16x4_f32(
                           false, aP, false, bV, (short)0, O[t], false, false);
            }
        }
    }

    // ---- export per-wave partials to LDS ----------------------------------
    if (ci == 0) {
#pragma unroll
        for (int r = 0; r < 8; ++r) {
            sM[wid][r + 8 * half] = mrow[r];
            sL[wid][r + 8 * half] = lrow[r];
        }
    }
#pragma unroll
    for (int t = 0; t < 8; ++t)
#pragma unroll
        for (int r = 0; r < 8; ++r)
            sO[wid][r + 8 * half][16 * t + ci] = O[t][r];
    __syncthreads();

    // ---- merge softmax stats across the 4 waves (16 threads) --------------
    if (threadIdx.x < 16) {
        const int n = threadIdx.x;
        const float m0s = sM[0][n], m1s = sM[1][n], m2s = sM[2][n], m3s = sM[3][n];
        const float ms  = fmaxf(fmaxf(m0s, m1s), fmaxf(m2s, m3s));
        const float s0 = __builtin_amdgcn_exp2f(m0s - ms);
        const float s1 = __builtin_amdgcn_exp2f(m1s - ms);
        const float s2 = __builtin_amdgcn_exp2f(m2s - ms);
        const float s3 = __builtin_amdgcn_exp2f(m3s - ms);
        sScale[0][n] = s0; sScale[1][n] = s1; sScale[2][n] = s2; sScale[3][n] = s3;
        sLinv[n] = 1.f / (sL[0][n] * s0 + sL[1][n] * s1 + sL[2][n] * s2 + sL[3][n] * s3);
    }
    __syncthreads();

    // ---- combine O partials, store partial output -------------------------
    // thread -> (n = tid&15, c8 = tid>>4); consecutive tid => consecutive n
    // in memory (coalesced 16-float runs per channel).
    {
        const int n  = threadIdx.x & 15;
        const int c8 = threadIdx.x >> 4;
        const float s0 = sScale[0][n], s1 = sScale[1][n];
        const float s2 = sScale[2][n], s3 = sScale[3][n];
        const float li = sLinv[n];
#pragma unroll
        for (int cb = 0; cb < 16; ++cb) {
            const int c = cb * 8 + c8;
            const float v = sO[0][n][c] * s0 + sO[1][n][c] * s1 +
                            sO[2][n][c] * s2 + sO[3][n][c] * s3;
            pb[(size_t)c * N_ + n0 + n] = v * li;
        }
    }
}

// ---------------------------------------------------------------------------
// out = y + sum_l gammas[l] * partial_l
// ---------------------------------------------------------------------------
__global__ __launch_bounds__(256) void final_add_kernel(
    const float* __restrict__ y, const float* __restrict__ gammas,
    const float* __restrict__ pOut, float* __restrict__ out)
{
    const int g = blockIdx.x * blockDim.x + threadIdx.x;    // 0 .. B*C*N-1
    float acc = y[g];
#pragma unroll
    for (int l = 0; l < L_; ++l)
        acc = fmaf(gammas[l], pOut[(size_t)l * B_ * C_ * N_ + g], acc);
    out[g] = acc;
}

// ---------------------------------------------------------------------------
extern "C" void kernel_launch(void* const* d_in, const int* in_sizes, int n_in,
                              void* d_out, int out_size, void* d_ws, size_t ws_size,
                              hipStream_t stream)
{
    (void)in_sizes; (void)n_in; (void)out_size; (void)ws_size;

    const float* stack  = (const float*)d_in[0];
    const float* y      = (const float*)d_in[1];
    const float* wq     = (const float*)d_in[2];
    const float* bq     = (const float*)d_in[3];
    const float* wk     = (const float*)d_in[4];
    const float* bk     = (const float*)d_in[5];
    const float* wv     = (const float*)d_in[6];
    const float* bv     = (const float*)d_in[7];
    const float* gammas = (const float*)d_in[8];
    float*       out    = (float*)d_out;

    // ws: qT [L*B*N*CQ] | k [L*B*CQ*N] | v [B*C*N] | partial [L*B*C*N]
    //   = 294912 + 294912 + 589824 + 2359296 floats = 14.2 MB
    float* ws   = (float*)d_ws;
    float* qT   = ws;
    float* kK   = qT + (size_t)L_ * B_ * N_ * CQ_;
    float* vV   = kK + (size_t)L_ * B_ * CQ_ * N_;
    float* pOut = vV + (size_t)B_ * C_ * N_;

    proj_qk_kernel<<<(L_ * B_ * N_) / 256, 256, 0, stream>>>(
        stack, wq, bq, wk, bk, qT, kK);
    proj_v_kernel<<<(B_ * C_ * N_) / 256, 256, 0, stream>>>(y, wv, bv, vV);

    flash_attn_kernel<<<L_ * B_ * NBLK, 128, 0, stream>>>(qT, kK, vV, pOut);

    final_add_kernel<<<(B_ * C_ * N_) / 256, 256, 0, stream>>>(
        y, gammas, pOut, out);
}